// MultiBoxLoss_40441412059294
// MI455X (gfx1250) — compile-verified
//
#include <hip/hip_runtime.h>
#include <math.h>

#define ALPHA_C 0.25f
#define FOUR_OVER_PI2 0.40528473456935108578f  // 4/pi^2

typedef __attribute__((ext_vector_type(2))) float v2f;
typedef __attribute__((ext_vector_type(8))) float v8f;

// ---------------------------------------------------------------------------
// Wave-wide f32 sum using V_WMMA_F32_16X16X4_F32 (CDNA5).
// A (16x4): lane L (L<16) -> A[L,0]=x, A[L,1]=0 ; lane L>=16 -> A[L-16,2]=x,
// A[L-16,3]=0.  B = all ones (layout-independent).  D[m,n] = x_m + x_{m+16}.
// Lane L holds D[r + 8*(L>=16), L%16] in c[r]; summing c[0..7] gives half the
// column, and shfl_xor(16) adds the other half -> every lane has the total.
// EXEC must be all-ones at the call site (callers guarantee this).
// ---------------------------------------------------------------------------
__device__ __forceinline__ float wave_sum_wmma(float x) {
  v2f a; a.x = x;    a.y = 0.0f;
  v2f b; b.x = 1.0f; b.y = 1.0f;
  v8f c = {};
  c = __builtin_amdgcn_wmma_f32_16x16x4_f32(
      /*neg_a=*/false, a, /*neg_b=*/false, b,
      /*c_mod=*/(short)0, c, /*reuse_a=*/false, /*reuse_b=*/false);
  float s = c[0] + c[1] + c[2] + c[3] + c[4] + c[5] + c[6] + c[7];
  s += __shfl_xor(s, 16, 32);
  return s;
}

__device__ __forceinline__ float clampf(float v, float lo, float hi) {
  return fminf(fmaxf(v, lo), hi);
}

// focal term for target==0 and target==1 on an already-clipped probability
__device__ __forceinline__ float focal_neg(float p) {
  return (1.0f - ALPHA_C) * p * p * (-logf(1.0f - p));
}
__device__ __forceinline__ float focal_pos(float p) {
  float q = 1.0f - p;
  return ALPHA_C * q * q * (-logf(p));
}

// ---------------------------------------------------------------------------
// Kernel 0: zero the per-batch accumulators in workspace.
// ws[0..B)    : num_pos per batch
// ws[B..2B)   : cls focal-loss sum per batch
// ws[2B..3B)  : reg ciou-loss sum per batch
// ---------------------------------------------------------------------------
__global__ void k_init(float* __restrict__ ws, int n) {
  int i = blockIdx.x * blockDim.x + threadIdx.x;
  if (i < n) ws[i] = 0.0f;
}

// ---------------------------------------------------------------------------
// Kernel 1: anchor->gt assignment (argmax IoU over Nb boxes, staged in LDS),
// CIoU reg loss for positive anchors, and the focal-loss correction term for
// the assigned class (f1(p)-f0(p)); wave-reduced via WMMA, atomically added.
// grid = (ceil(Na/256), B), block = 256.
// ---------------------------------------------------------------------------
#define MAX_NB 128

__global__ void k_assign(const float* __restrict__ cls_pred,
                         const float* __restrict__ reg_pred,
                         const float* __restrict__ annots,
                         const float* __restrict__ anchors,
                         float* __restrict__ ws,
                         int B, int C, int Na, int Nb) {
  __shared__ __attribute__((aligned(16))) float sbox[MAX_NB * 5];

  const int b   = blockIdx.y;
  const int tid = threadIdx.x;
  const int a   = blockIdx.x * blockDim.x + tid;
  const int nb5 = Nb * 5;
  const float* ab = annots + (size_t)b * nb5;

  // Stage this batch's annotations into LDS (async DMA path on gfx1250).
#if defined(__gfx1250__) && __has_builtin(__builtin_amdgcn_global_load_async_to_lds_b32)
  for (int i = tid; i < nb5; i += blockDim.x) {
    __builtin_amdgcn_global_load_async_to_lds_b32(
        (__attribute__((address_space(1))) int*)(void*)(ab + i),
        (__attribute__((address_space(3))) int*)(void*)&sbox[i],
        0, 0);
  }
#if __has_builtin(__builtin_amdgcn_s_wait_asynccnt)
  __builtin_amdgcn_s_wait_asynccnt(0);
#else
  asm volatile("s_wait_asynccnt 0" ::: "memory");
#endif
#else
  for (int i = tid; i < nb5; i += blockDim.x) sbox[i] = ab[i];
#endif
  __syncthreads();

  float posc = 0.0f, ciou_l = 0.0f, corr = 0.0f;

  if (a < Na) {
    const float4 A4 = ((const float4*)anchors)[a];
    const float aw = A4.z - A4.x, ah = A4.w - A4.y;
    const float acx = A4.x + 0.5f * aw, acy = A4.y + 0.5f * ah;
    const float a_area = aw * ah;

    float best = -1.0f;
    int bestj = 0;
    for (int j = 0; j < Nb; ++j) {
      const float bx1 = sbox[j * 5 + 0];
      const float by1 = sbox[j * 5 + 1];
      const float bx2 = sbox[j * 5 + 2];
      const float by2 = sbox[j * 5 + 3];
      const float lab = sbox[j * 5 + 4];
      const float barea = (bx2 - bx1) * (by2 - by1);
      float iw = fminf(A4.z, bx2) - fmaxf(A4.x, bx1);
      float ih = fminf(A4.w, by2) - fmaxf(A4.y, by1);
      iw = fmaxf(iw, 0.0f);
      ih = fmaxf(ih, 0.0f);
      const float inter = iw * ih;
      const float uni = fmaxf(a_area + barea - inter, 1e-7f);
      float iou = inter / uni;
      iou = (lab != -1.0f) ? iou : -1.0f;
      if (iou > best) { best = iou; bestj = j; }  // first-max like argmax
    }

    if (best >= 0.5f) {
      posc = 1.0f;
      const float x1 = sbox[bestj * 5 + 0];
      const float y1 = sbox[bestj * 5 + 1];
      const float x2 = sbox[bestj * 5 + 2];
      const float y2 = sbox[bestj * 5 + 3];
      const int label = (int)sbox[bestj * 5 + 4];

      const float gw = fmaxf(x2 - x1, 1.0f);
      const float gh = fmaxf(y2 - y1, 1.0f);
      const float gcx = x1 + 0.5f * gw;
      const float gcy = y1 + 0.5f * gh;

      // regression: reg_pred[b, k, a] * STD
      const size_t rbase = (size_t)b * 4 * (size_t)Na + (size_t)a;
      const float r0 = reg_pred[rbase + 0 * (size_t)Na] * 0.1f;
      const float r1 = reg_pred[rbase + 1 * (size_t)Na] * 0.1f;
      const float r2 = reg_pred[rbase + 2 * (size_t)Na] * 0.2f;
      const float r3 = reg_pred[rbase + 3 * (size_t)Na] * 0.2f;

      const float pcx = acx + r0 * aw;
      const float pcy = acy + r1 * ah;
      const float pw = expf(r2) * aw;
      const float ph = expf(r3) * ah;

      const float g_area = gw * gh, p_area = pw * ph;
      const float ix1 = fmaxf(gcx - 0.5f * gw, pcx - 0.5f * pw);
      const float ix2 = fminf(gcx + 0.5f * gw, pcx + 0.5f * pw);
      const float iy1 = fmaxf(gcy - 0.5f * gh, pcy - 0.5f * ph);
      const float iy2 = fminf(gcy + 0.5f * gh, pcy + 0.5f * ph);
      const float inter = fmaxf(ix2 - ix1, 0.0f) * fmaxf(iy2 - iy1, 0.0f);
      const float ex1 = fminf(gcx - 0.5f * gw, pcx - 0.5f * pw);
      const float ex2 = fmaxf(gcx + 0.5f * gw, pcx + 0.5f * pw);
      const float ey1 = fminf(gcy - 0.5f * gh, pcy - 0.5f * ph);
      const float ey2 = fmaxf(gcy + 0.5f * gh, pcy + 0.5f * ph);
      const float dcx = gcx - pcx, dcy = gcy - pcy;
      const float idiag = dcx * dcx + dcy * dcy;
      const float ew = fmaxf(ex2 - ex1, 0.0f), eh = fmaxf(ey2 - ey1, 0.0f);
      const float ediag = ew * ew + eh * eh;
      const float uni = g_area + p_area - inter;
      const float u = idiag / fmaxf(ediag, 1e-6f);
      const float iou = inter / fmaxf(uni, 1e-6f);
      const float dat = atanf(gw / gh) - atanf(pw / ph);
      const float v = FOUR_OVER_PI2 * dat * dat;
      const float S = (iou > 0.5f) ? 1.0f : 0.0f;
      const float alp = S * v / fmaxf(1.0f - iou + v, 1e-6f);
      const float ci = clampf(iou - u - alp * v, -1.0f, 1.0f);
      ciou_l = 1.0f - ci;

      // focal correction for the assigned class (target flips 0 -> 1)
      float p = cls_pred[((size_t)b * C + (size_t)label) * (size_t)Na + (size_t)a];
      p = clampf(p, 1e-7f, 1.0f - 1e-7f);
      corr = focal_pos(p) - focal_neg(p);
    }
  }

  // Full-EXEC wave reductions via WMMA, then one atomic per wave per value.
  const float sp = wave_sum_wmma(posc);
  const float sr = wave_sum_wmma(ciou_l);
  const float sc = wave_sum_wmma(corr);
  if ((tid & 31) == 0) {
    atomicAdd(&ws[b], sp);
    atomicAdd(&ws[2 * B + b], sr);
    atomicAdd(&ws[B + b], sc);
  }
}

// ---------------------------------------------------------------------------
// Kernel 2: streaming focal-loss (target==0 branch) over all of cls_pred.
// Each thread owns 4 consecutive anchors (float4 / B128 loads) and loops over
// the C classes; wave sums via WMMA; one atomic per wave.
// grid = (ceil(Na/4/256), B), block = 256.
// ---------------------------------------------------------------------------
__global__ void k_focal(const float* __restrict__ cls_pred,
                        float* __restrict__ ws,
                        int B, int C, int Na) {
  const int b  = blockIdx.y;
  const int t  = blockIdx.x * blockDim.x + threadIdx.x;
  const int a4 = t * 4;
  float s = 0.0f;

  const float* base = cls_pred + (size_t)b * (size_t)C * (size_t)Na;
  if (a4 + 3 < Na) {
#pragma unroll 4
    for (int c = 0; c < C; ++c) {
      const float4 p4 = *(const float4*)(base + (size_t)c * (size_t)Na + a4);
      float p;
      p = clampf(p4.x, 1e-7f, 1.0f - 1e-7f); s += focal_neg(p);
      p = clampf(p4.y, 1e-7f, 1.0f - 1e-7f); s += focal_neg(p);
      p = clampf(p4.z, 1e-7f, 1.0f - 1e-7f); s += focal_neg(p);
      p = clampf(p4.w, 1e-7f, 1.0f - 1e-7f); s += focal_neg(p);
    }
  } else if (a4 < Na) {
    for (int c = 0; c < C; ++c) {
      for (int k = 0; k < 4 && a4 + k < Na; ++k) {
        float p = base[(size_t)c * (size_t)Na + a4 + k];
        p = clampf(p, 1e-7f, 1.0f - 1e-7f);
        s += focal_neg(p);
      }
    }
  }

  const float ssum = wave_sum_wmma(s);
  if ((threadIdx.x & 31) == 0) atomicAdd(&ws[B + b], ssum);
}

// ---------------------------------------------------------------------------
// Kernel 3: finalize -> out[0] = mean_b cls_loss, out[1] = mean_b reg_loss
// ---------------------------------------------------------------------------
__global__ void k_final(const float* __restrict__ ws, float* __restrict__ out, int B) {
  if (threadIdx.x == 0 && blockIdx.x == 0) {
    float cls_m = 0.0f, reg_m = 0.0f;
    for (int b = 0; b < B; ++b) {
      const float np = fmaxf(ws[b], 1.0f);
      cls_m += ws[B + b] / np;
      const float rl = ws[2 * B + b] / np;
      reg_m += (ws[b] > 0.0f) ? rl : 0.0f;
    }
    out[0] = cls_m / (float)B;
    out[1] = reg_m / (float)B;
  }
}

extern "C" void kernel_launch(void* const* d_in, const int* in_sizes, int n_in,
                              void* d_out, int out_size, void* d_ws, size_t ws_size,
                              hipStream_t stream) {
  const float* cls_pred = (const float*)d_in[0];
  const float* reg_pred = (const float*)d_in[1];
  const float* annots   = (const float*)d_in[2];
  const float* anchors  = (const float*)d_in[3];
  float* out = (float*)d_out;
  float* ws  = (float*)d_ws;

  const int Na = in_sizes[3] / 4;               // anchors: (Na, 4)
  const int B  = in_sizes[1] / (4 * Na);        // reg_pred: (B, 4, Na)
  const int C  = in_sizes[0] / (B * Na);        // cls_pred: (B, C, Na)
  int Nb = in_sizes[2] / (B * 5);               // annots: (B, Nb, 5)
  if (Nb > MAX_NB) Nb = MAX_NB;                 // LDS stage capacity (ref: 64)

  k_init<<<1, 64, 0, stream>>>(ws, 3 * B);

  dim3 gridA((Na + 255) / 256, B);
  k_assign<<<gridA, 256, 0, stream>>>(cls_pred, reg_pred, annots, anchors, ws,
                                      B, C, Na, Nb);

  const int na4 = (Na + 3) / 4;
  dim3 gridF((na4 + 255) / 256, B);
  k_focal<<<gridF, 256, 0, stream>>>(cls_pred, ws, B, C, Na);

  k_final<<<1, 32, 0, stream>>>(ws, out, B);
}